// GATLayer_6468220748676
// MI455X (gfx1250) — compile-verified
//
#include <hip/hip_runtime.h>
#include <math.h>

typedef __attribute__((ext_vector_type(2))) float v2f;
typedef __attribute__((ext_vector_type(8))) float v8f;

#define IN_DIM 128
#define OUT_DIM 128

// ---------------------------------------------------------------------------
// Init: h = 0 (output is poisoned by harness), m = -inf (segment_max identity),
// denom = 0.
// ---------------------------------------------------------------------------
__global__ void gat_init_kernel(float* __restrict__ h, float* __restrict__ mmax,
                                float* __restrict__ denom, int n_nodes) {
    int i = blockIdx.x * blockDim.x + threadIdx.x;
    int total = n_nodes * OUT_DIM;
    if (i < total) h[i] = 0.0f;
    if (i < n_nodes) {
        mmax[i]  = -INFINITY;
        denom[i] = 0.0f;
    }
}

// ---------------------------------------------------------------------------
// z = X @ W^T via V_WMMA_F32_16X16X4_F32 (fp32 exact, matches reference).
// One wave32 per 16x16 tile of z. Block = 256 threads = 8 waves covering the
// full 128-wide N dimension for one 16-row M slab; grid = n_nodes/16 blocks.
//
// A fragment (16x4 f32, 2 VGPRs):  lanes 0-15 hold (M=lane, K=k,k+1),
//                                  lanes 16-31 hold (M=lane-16, K=k+2,k+3).
// B fragment (4x16 f32, 2 VGPRs):  mirrored: B[k][n] = W[n*IN_DIM + k], so the
//                                  per-lane data is a contiguous float2 of W.
// C/D (16x16 f32, 8 VGPRs): vgpr r: lanes0-15 -> (M=r, N=lane),
//                                    lanes16-31 -> (M=8+r, N=lane-16).
// ---------------------------------------------------------------------------
__global__ void gat_gemm_wmma(const float* __restrict__ X, const float* __restrict__ W,
                              float* __restrict__ z, int n_nodes) {
    const int lane = threadIdx.x & 31;
    const int wave = threadIdx.x >> 5;
    const int m0 = blockIdx.x * 16;
    const int n0 = wave * 16;
    const int ml = lane & 15;   // row (A) / col (B,C,D) within tile
    const int hi = lane >> 4;   // selects K pair 0,1 vs 2,3; row +8 for C/D

    const float* Arow = X + (size_t)(m0 + ml) * IN_DIM + 2 * hi;
    const float* Brow = W + (size_t)(n0 + ml) * IN_DIM + 2 * hi;

    v8f c = {};
    #pragma unroll 4
    for (int k = 0; k < IN_DIM; k += 4) {
        v2f a = *(const v2f*)(Arow + k);
        v2f b = *(const v2f*)(Brow + k);
        c = __builtin_amdgcn_wmma_f32_16x16x4_f32(
                /*neg_a=*/false, a, /*neg_b=*/false, b,
                /*c_mod=*/(short)0, c, /*reuse_a=*/false, /*reuse_b=*/false);
    }

    float* zb = z + (size_t)m0 * OUT_DIM + n0 + ml;
    #pragma unroll
    for (int r = 0; r < 8; ++r) {
        zb[(size_t)(r + hi * 8) * OUT_DIM] = c[r];
    }
}

// ---------------------------------------------------------------------------
// Per-node attention scores: s_l = z . a_l, s_r = z . a_r. One wave per node,
// float4 per lane, wave32 xor-shuffle reduction.
// ---------------------------------------------------------------------------
__global__ void gat_scores_kernel(const float* __restrict__ z, const float* __restrict__ a_attn,
                                  float* __restrict__ s_l, float* __restrict__ s_r, int n_nodes) {
    int gw   = (blockIdx.x * blockDim.x + threadIdx.x) >> 5;
    int lane = threadIdx.x & 31;
    if (gw >= n_nodes) return;

    const float4 zv = ((const float4*)(z + (size_t)gw * OUT_DIM))[lane];
    const float4 al = ((const float4*)a_attn)[lane];
    const float4 ar = ((const float4*)(a_attn + OUT_DIM))[lane];

    float sl = zv.x * al.x + zv.y * al.y + zv.z * al.z + zv.w * al.w;
    float sr = zv.x * ar.x + zv.y * ar.y + zv.z * ar.z + zv.w * ar.w;

    #pragma unroll
    for (int off = 16; off > 0; off >>= 1) {
        sl += __shfl_xor(sl, off, 32);
        sr += __shfl_xor(sr, off, 32);
    }
    if (lane == 0) { s_l[gw] = sl; s_r[gw] = sr; }
}

// Hardware atomic float-max via signed/unsigned integer atomics (valid with
// -inf initialization): nonneg floats compare as signed ints, negative floats
// compare reversed as unsigned ints.
__device__ __forceinline__ void atomicMaxF(float* addr, float val) {
    if (val >= 0.0f)
        atomicMax((int*)addr, __float_as_int(val));
    else
        atomicMin((unsigned int*)addr, __float_as_uint(val));
}

// ---------------------------------------------------------------------------
// Per-edge: e = leaky_relu(s_l[src] + s_r[dst], 0.01); segment max into m[dst].
// ---------------------------------------------------------------------------
__global__ void gat_edge_max_kernel(const float* __restrict__ s_l, const float* __restrict__ s_r,
                                    const int* __restrict__ src, const int* __restrict__ dst,
                                    float* __restrict__ e_buf, float* __restrict__ mmax, int n_edges) {
    int i = blockIdx.x * blockDim.x + threadIdx.x;
    if (i >= n_edges) return;
    float e = s_l[src[i]] + s_r[dst[i]];
    e = (e > 0.0f) ? e : 0.01f * e;
    e_buf[i] = e;
    atomicMaxF(&mmax[dst[i]], e);
}

// ---------------------------------------------------------------------------
// Per-edge: ex = exp(e - m[dst]); segment sum into denom[dst].
// ---------------------------------------------------------------------------
__global__ void gat_edge_exp_kernel(const float* __restrict__ e_buf, const int* __restrict__ dst,
                                    const float* __restrict__ mmax, float* __restrict__ ex_buf,
                                    float* __restrict__ denom, int n_edges) {
    int i = blockIdx.x * blockDim.x + threadIdx.x;
    if (i >= n_edges) return;
    int d = dst[i];
    float ex = expf(e_buf[i] - mmax[d]);
    ex_buf[i] = ex;
    unsafeAtomicAdd(&denom[d], ex);
}

// ---------------------------------------------------------------------------
// Aggregation (the bandwidth-dominant phase): one wave per edge.
// alpha = ex/denom[dst]; h[dst] += alpha * z[src]  (float4 gather per lane,
// 4x global_atomic_add_f32 per lane; h and z live in the 192MB L2).
// ---------------------------------------------------------------------------
__global__ void gat_aggregate_kernel(const float* __restrict__ z, const float* __restrict__ ex_buf,
                                     const float* __restrict__ denom,
                                     const int* __restrict__ src, const int* __restrict__ dst,
                                     float* __restrict__ h, int n_edges) {
    int gw   = (blockIdx.x * blockDim.x + threadIdx.x) >> 5;
    int lane = threadIdx.x & 31;
    if (gw >= n_edges) return;

    int s = src[gw];
    int d = dst[gw];
    float alpha = ex_buf[gw] / denom[d];

    const float4 zv = ((const float4*)(z + (size_t)s * OUT_DIM))[lane];
    float* hb = h + (size_t)d * OUT_DIM + lane * 4;
    unsafeAtomicAdd(hb + 0, alpha * zv.x);
    unsafeAtomicAdd(hb + 1, alpha * zv.y);
    unsafeAtomicAdd(hb + 2, alpha * zv.z);
    unsafeAtomicAdd(hb + 3, alpha * zv.w);
}

extern "C" void kernel_launch(void* const* d_in, const int* in_sizes, int n_in,
                              void* d_out, int out_size, void* d_ws, size_t ws_size,
                              hipStream_t stream) {
    const float* features = (const float*)d_in[0];
    const int*   src_idx  = (const int*)d_in[1];
    const int*   dst_idx  = (const int*)d_in[2];
    const float* W_fc     = (const float*)d_in[3];
    const float* a_attn   = (const float*)d_in[4];
    float* h = (float*)d_out;

    const int n_nodes = in_sizes[0] / IN_DIM;   // 40000
    const int n_edges = in_sizes[1];            // 640000

    // Workspace layout (~26 MB)
    float* z      = (float*)d_ws;                      // n_nodes * 128
    float* s_l    = z    + (size_t)n_nodes * OUT_DIM;  // n_nodes
    float* s_r    = s_l  + n_nodes;                    // n_nodes
    float* mmax   = s_r  + n_nodes;                    // n_nodes
    float* denom  = mmax + n_nodes;                    // n_nodes
    float* e_buf  = denom + n_nodes;                   // n_edges
    float* ex_buf = e_buf + n_edges;                   // n_edges

    gat_init_kernel<<<(n_nodes * OUT_DIM + 255) / 256, 256, 0, stream>>>(h, mmax, denom, n_nodes);
    gat_gemm_wmma<<<(n_nodes + 15) / 16, 256, 0, stream>>>(features, W_fc, z, n_nodes);
    gat_scores_kernel<<<(n_nodes + 7) / 8, 256, 0, stream>>>(z, a_attn, s_l, s_r, n_nodes);
    gat_edge_max_kernel<<<(n_edges + 255) / 256, 256, 0, stream>>>(s_l, s_r, src_idx, dst_idx,
                                                                   e_buf, mmax, n_edges);
    gat_edge_exp_kernel<<<(n_edges + 255) / 256, 256, 0, stream>>>(e_buf, dst_idx, mmax,
                                                                   ex_buf, denom, n_edges);
    gat_aggregate_kernel<<<(n_edges + 7) / 8, 256, 0, stream>>>(z, ex_buf, denom,
                                                                src_idx, dst_idx, h, n_edges);
}